// LoRaMHA_42623255445966
// MI455X (gfx1250) — compile-verified
//
#include <hip/hip_runtime.h>
#include <hip/hip_bf16.h>

// ---------------- problem constants (from reference) ----------------
#define BB    2
#define SS    2048
#define DD    1024
#define HH    16
#define HD    64
#define RANK  8
#define ALPHA 8.0f
#define NEGV  (-1e9f)

typedef __attribute__((ext_vector_type(2))) float v2f;
typedef __attribute__((ext_vector_type(8))) float v8f;

// D = A(16x4 f32) * B(4x16 f32) + C(16x16 f32) -- exact fp32 matrix op
__device__ __forceinline__ v8f wmma4(v2f a, v2f b, v8f c) {
    return __builtin_amdgcn_wmma_f32_16x16x4_f32(
        /*neg_a=*/false, a, /*neg_b=*/false, b,
        /*c_mod=*/(short)0, c, /*reuse_a=*/false, /*reuse_b=*/false);
}

// ---------------- async global->LDS staging (guarded; falls back to ds stores) ----
#if defined(__gfx1250__) && __has_builtin(__builtin_amdgcn_global_load_async_to_lds_b128)
#define HAVE_ASYNC_COPY 1
// Parameter types per hipcc diagnostic: int __vector_size__(16), AS1 / AS3.
typedef int vsi4 __attribute__((__vector_size__(4 * sizeof(int))));
typedef __attribute__((address_space(1))) vsi4* gvec4p;
typedef __attribute__((address_space(3))) vsi4* lvec4p;

__device__ __forceinline__ void async_cp16(const float* g, float* l) {
    // ptr -> int -> AS-qualified ptr: AS1 == flat global; AS3 takes addr[31:0]
    // which is exactly the ISA's flat->LDS aperture mapping (truncate).
    __builtin_amdgcn_global_load_async_to_lds_b128(
        (gvec4p)(unsigned long long)(__UINTPTR_TYPE__)g,
        (lvec4p)(unsigned int)(__UINTPTR_TYPE__)l,
        /*offset=*/0, /*cpol=*/0);
}
#if __has_builtin(__builtin_amdgcn_s_wait_asynccnt)
#define WAIT_ASYNC() __builtin_amdgcn_s_wait_asynccnt(0)
#else
#define WAIT_ASYNC() asm volatile("s_wait_asynccnt 0x0" ::: "memory")
#endif
#else
#define HAVE_ASYNC_COPY 0
#define WAIT_ASYNC()
#endif

// ---------------- kernel 1: xa = x @ A_lora  [4096 x 8] ----------------
__global__ __launch_bounds__(256)
void lora_xa_kernel(const float* __restrict__ x,
                    const float* __restrict__ A,
                    float* __restrict__ xa) {
    int idx = blockIdx.x * 256 + threadIdx.x;      // exactly 4096*8 threads
    int row = idx >> 3;
    int r   = idx & 7;
    const float* xr = x + (size_t)row * DD;
    float acc = 0.0f;
#pragma unroll 8
    for (int k = 0; k < DD; ++k)
        acc += xr[k] * A[k * RANK + r];
    xa[idx] = acc;
}

// ---------------- kernel 2/4: Out = X@W + bias (+ ALPHA * XA@BL) ----------------
// X: MxK (row-major, lda=K), W: KxN (row-major, ldb=N), Out: MxN
// 64x64 workgroup tile, 256 threads = 8 waves, each wave -> 16x32 strip.
__global__ __launch_bounds__(256)
void gemm_wmma_kernel(const float* __restrict__ X,
                      const float* __restrict__ W,
                      const float* __restrict__ bias,
                      const float* __restrict__ XA,   // M x RANK or nullptr
                      const float* __restrict__ BL,   // RANK x N or nullptr
                      float* __restrict__ Out,
                      int M, int N, int K) {
    // row strides are 16B multiples (B128 LDS writes) and bank-conflict-free
    __shared__ __align__(16) float Xs[64][36];
    __shared__ __align__(16) float Ws[32][68];

    const int rowBase = blockIdx.y * 64;
    const int colBase = blockIdx.x * 64;

    const int tid  = threadIdx.x;
    const int wave = tid >> 5;         // 0..7
    const int lane = tid & 31;
    const int hl   = lane >> 4;        // half-wave select (K pair / M+8)
    const int lm   = lane & 15;
    const int wm   = wave & 3;         // 4 waves along M
    const int wn   = wave >> 2;        // 2 waves along N

    v8f acc0 = {};                     // n offsets [0,16)
    v8f acc1 = {};                     // n offsets [16,32)

    for (int kb = 0; kb < K; kb += 32) {
        __syncthreads();               // previous tile fully consumed
#if HAVE_ASYNC_COPY
        // X tile 64x32 = 512 x 16B chunks, W tile 32x64 = 512 x 16B chunks
#pragma unroll
        for (int j = 0; j < 2; ++j) {
            int chunk = tid + j * 256;
            int xr = chunk >> 3, xc = (chunk & 7) << 2;      // 8 chunks / X row
            async_cp16(&X[(size_t)(rowBase + xr) * K + kb + xc], &Xs[xr][xc]);
            int wr = chunk >> 4, wc = (chunk & 15) << 2;     // 16 chunks / W row
            async_cp16(&W[(size_t)(kb + wr) * N + colBase + wc], &Ws[wr][wc]);
        }
        WAIT_ASYNC();
#else
        for (int i = tid; i < 64 * 32; i += 256) {
            int r = i >> 5, c = i & 31;
            Xs[r][c] = X[(size_t)(rowBase + r) * K + kb + c];
        }
        for (int i = tid; i < 32 * 64; i += 256) {
            int r = i >> 6, c = i & 63;
            Ws[r][c] = W[(size_t)(kb + r) * N + colBase + c];
        }
#endif
        if (kb + 32 < K) {   // hint next tiles into cache (global_prefetch_b8)
            __builtin_prefetch(&X[(size_t)(rowBase + (tid >> 2)) * K + kb + 32 + ((tid & 3) << 3)], 0, 1);
            __builtin_prefetch(&W[(size_t)(kb + 32 + (tid >> 3)) * N + colBase + ((tid & 7) << 3)], 0, 1);
        }
        __syncthreads();

#pragma unroll
        for (int k0 = 0; k0 < 32; k0 += 4) {
            v2f a, b0, b1;
            a.x  = Xs[wm * 16 + lm][k0 + 2 * hl];
            a.y  = Xs[wm * 16 + lm][k0 + 2 * hl + 1];
            b0.x = Ws[k0 + 2 * hl][wn * 32 + lm];
            b0.y = Ws[k0 + 2 * hl + 1][wn * 32 + lm];
            b1.x = Ws[k0 + 2 * hl][wn * 32 + 16 + lm];
            b1.y = Ws[k0 + 2 * hl + 1][wn * 32 + 16 + lm];
            acc0 = wmma4(a, b0, acc0);
            acc1 = wmma4(a, b1, acc1);
        }
    }

    // epilogue: bias + optional LoRA correction, then store
    const int row0 = rowBase + wm * 16;
    const int col0 = colBase + wn * 32 + lm;
#pragma unroll
    for (int e = 0; e < 8; ++e) {
        int row = row0 + e + 8 * hl;           // C/D layout: M = e + 8*half
        float s0 = acc0[e] + bias[col0];
        float s1 = acc1[e] + bias[col0 + 16];
        if (XA != nullptr) {
            float l0 = 0.0f, l1 = 0.0f;
#pragma unroll
            for (int r = 0; r < RANK; ++r) {
                float xav = XA[row * RANK + r];
                l0 += xav * BL[r * N + col0];
                l1 += xav * BL[r * N + col0 + 16];
            }
            s0 += ALPHA * l0;
            s1 += ALPHA * l1;
        }
        Out[(size_t)row * N + col0]      = s0;
        Out[(size_t)row * N + col0 + 16] = s1;
    }
}

// ---------------- kernel 3: causal flash attention ----------------
// qkv: [B*S, 3072] with q|k|v each D wide; ctx out: [B*S, D] heads merged.
// grid: (S/64, H, B); block: 128 threads = 4 waves, each wave owns 16 queries.
__global__ __launch_bounds__(128)
void attn_kernel(const float* __restrict__ qkv, float* __restrict__ ctx) {
    __shared__ __align__(16) float Ks[16][68];
    __shared__ __align__(16) float Vs[16][68];
    __shared__ __align__(16) float Ps[4][16][20];

    const int b     = blockIdx.z;
    const int h     = blockIdx.y;
    const int qBase = blockIdx.x * 64;

    const int tid  = threadIdx.x;
    const int wave = tid >> 5;        // 0..3
    const int lane = tid & 31;
    const int hl   = lane >> 4;
    const int lm   = lane & 15;

    // preload this lane's Q fragments (row = lm of the wave's 16-query strip)
    const int qRow = qBase + wave * 16 + lm;
    const float* qp = qkv + (size_t)(b * SS + qRow) * 3072 + h * HD;
    float qreg[32];
#pragma unroll
    for (int i = 0; i < 16; ++i) {
        qreg[2 * i]     = qp[4 * i + 2 * hl];
        qreg[2 * i + 1] = qp[4 * i + 2 * hl + 1];
    }

    v8f o0 = {}, o1 = {}, o2 = {}, o3 = {};
    float mrow[8], lrow[8];
#pragma unroll
    for (int e = 0; e < 8; ++e) { mrow[e] = -3.0e38f; lrow[e] = 0.0f; }

    const int nKeyTiles = (qBase >> 4) + 4;     // causal: keys <= qBase+63
    for (int kt = 0; kt < nKeyTiles; ++kt) {
        const int keyBase = kt * 16;
        __syncthreads();
#if HAVE_ASYNC_COPY
        // K tile and V tile: 16x64 floats = 256 x 16B chunks each
#pragma unroll
        for (int j = 0; j < 2; ++j) {
            int chunk = tid + j * 128;
            int r = chunk >> 4, c = (chunk & 15) << 2;       // 16 chunks / row
            size_t ro = (size_t)(b * SS + keyBase + r) * 3072 + h * HD + c;
            async_cp16(&qkv[ro + DD],     &Ks[r][c]);
            async_cp16(&qkv[ro + 2 * DD], &Vs[r][c]);
        }
        WAIT_ASYNC();
#else
        for (int i = tid; i < 16 * 64; i += 128) {
            int r = i >> 6, c = i & 63;
            size_t ro = (size_t)(b * SS + keyBase + r) * 3072 + h * HD;
            Ks[r][c] = qkv[ro + DD + c];        // K block
            Vs[r][c] = qkv[ro + 2 * DD + c];    // V block
        }
#endif
        __syncthreads();

        // S = Q(16x64) * K^T(64x16)
        v8f s = {};
#pragma unroll
        for (int i = 0; i < 16; ++i) {
            v2f a, bb;
            a.x  = qreg[2 * i];
            a.y  = qreg[2 * i + 1];
            bb.x = Ks[lm][4 * i + 2 * hl];
            bb.y = Ks[lm][4 * i + 2 * hl + 1];
            s = wmma4(a, bb, s);
        }

        // mask-then-scale exactly like reference: (s + mask) / sqrt(HD)
        const int key = keyBase + lm;
        float tmax[8];
#pragma unroll
        for (int e = 0; e < 8; ++e) {
            int q = qBase + wave * 16 + e + 8 * hl;
            s[e] = (s[e] + ((key <= q) ? 0.0f : NEGV)) * 0.125f;
        }
        // row max across the 16-lane half-groups (rows live per half)
#pragma unroll
        for (int e = 0; e < 8; ++e) {
            float v = s[e];
            v = fmaxf(v, __shfl_xor(v, 8, 16));
            v = fmaxf(v, __shfl_xor(v, 4, 16));
            v = fmaxf(v, __shfl_xor(v, 2, 16));
            v = fmaxf(v, __shfl_xor(v, 1, 16));
            tmax[e] = v;
        }
        float scl[8];
#pragma unroll
        for (int e = 0; e < 8; ++e) {
            float mo = mrow[e];
            float mn = fmaxf(mo, tmax[e]);
            mrow[e]  = mn;
            scl[e]   = __expf(mo - mn);
            s[e]     = __expf(s[e] - mn);       // P
        }
#pragma unroll
        for (int e = 0; e < 8; ++e) {
            float v = s[e];
            v += __shfl_xor(v, 8, 16);
            v += __shfl_xor(v, 4, 16);
            v += __shfl_xor(v, 2, 16);
            v += __shfl_xor(v, 1, 16);
            lrow[e] = lrow[e] * scl[e] + v;
            o0[e] *= scl[e]; o1[e] *= scl[e]; o2[e] *= scl[e]; o3[e] *= scl[e];
        }

        // C-layout -> A-layout via LDS (wave-private region, in-order DS)
#pragma unroll
        for (int e = 0; e < 8; ++e)
            Ps[wave][e + 8 * hl][lm] = s[e];

        // O += P(16x16) * V(16x64)
#pragma unroll
        for (int i = 0; i < 4; ++i) {
            v2f a, b0, b1, b2, b3;
            a.x  = Ps[wave][lm][4 * i + 2 * hl];
            a.y  = Ps[wave][lm][4 * i + 2 * hl + 1];
            b0.x = Vs[4 * i + 2 * hl][lm];      b0.y = Vs[4 * i + 2 * hl + 1][lm];
            b1.x = Vs[4 * i + 2 * hl][16 + lm]; b1.y = Vs[4 * i + 2 * hl + 1][16 + lm];
            b2.x = Vs[4 * i + 2 * hl][32 + lm]; b2.y = Vs[4 * i + 2 * hl + 1][32 + lm];
            b3.x = Vs[4 * i + 2 * hl][48 + lm]; b3.y = Vs[4 * i + 2 * hl + 1][48 + lm];
            o0 = wmma4(a, b0, o0);
            o1 = wmma4(a, b1, o1);
            o2 = wmma4(a, b2, o2);
            o3 = wmma4(a, b3, o3);
        }
    }

    // normalize and store merged-head context [B*S, D]
    const size_t outBase = (size_t)(b * SS + qBase + wave * 16) * DD + h * HD;
#pragma unroll
    for (int e = 0; e < 8; ++e) {
        int r = e + 8 * hl;
        float inv = 1.0f / lrow[e];
        size_t ro = outBase + (size_t)r * DD;
        ctx[ro + lm]      = o0[e] * inv;
        ctx[ro + 16 + lm] = o1[e] * inv;
        ctx[ro + 32 + lm] = o2[e] * inv;
        ctx[ro + 48 + lm] = o3[e] * inv;
    }
}

// ---------------- host-side orchestration ----------------
extern "C" void kernel_launch(void* const* d_in, const int* in_sizes, int n_in,
                              void* d_out, int out_size, void* d_ws, size_t ws_size,
                              hipStream_t stream) {
    (void)in_sizes; (void)n_in; (void)out_size; (void)ws_size;

    const float* x      = (const float*)d_in[0];
    const float* W_qkv  = (const float*)d_in[1];
    const float* b_qkv  = (const float*)d_in[2];
    const float* A_lora = (const float*)d_in[3];
    const float* B_lora = (const float*)d_in[4];
    const float* W_proj = (const float*)d_in[5];
    const float* b_proj = (const float*)d_in[6];
    float* out = (float*)d_out;

    const int M = BB * SS;                       // 4096 rows
    char* ws = (char*)d_ws;
    float* qkv = (float*)ws;                                         // M x 3072
    float* ctx = (float*)(ws + (size_t)M * 3 * DD * sizeof(float));  // M x 1024
    float* xa  = (float*)(ws + (size_t)M * 4 * DD * sizeof(float));  // M x 8

    // 1) LoRA down-projection
    lora_xa_kernel<<<dim3((M * RANK) / 256), dim3(256), 0, stream>>>(x, A_lora, xa);
    // 2) fused qkv = x@W_qkv + b + ALPHA * xa@B_lora
    gemm_wmma_kernel<<<dim3((3 * DD) / 64, M / 64), dim3(256), 0, stream>>>(
        x, W_qkv, b_qkv, xa, B_lora, qkv, M, 3 * DD, DD);
    // 3) causal flash attention -> merged-head ctx
    attn_kernel<<<dim3(SS / 64, HH, BB), dim3(128), 0, stream>>>(qkv, ctx);
    // 4) output projection
    gemm_wmma_kernel<<<dim3(DD / 64, M / 64), dim3(256), 0, stream>>>(
        ctx, W_proj, b_proj, nullptr, nullptr, out, M, DD, DD);
}